// CoeffProtoAttention_56659208568929
// MI455X (gfx1250) — compile-verified
//
#include <hip/hip_runtime.h>

typedef __attribute__((ext_vector_type(2))) float v2f;
typedef __attribute__((ext_vector_type(8))) float v8f;

#define LN_EPS 1e-5f
#define NM 64
#define AA 8400
#define EE 128
#define TILES_PER_N 525              // 8400 / 16
#define N_TILES (NM * TILES_PER_N)   // 33600
#define SCALE_QK 0.17677669529663687f // 32^-0.5

// workspace layout (floats)
#define WS_POOLED 0      // [64]
#define WS_K      64     // Kw[256] Kb[256] Kc[256] Vo[256]  (h*64+m)
#define WS_STATS  1088   // Vw, Cwb, Vb

// ---- DPP 16-lane butterfly helpers (stay within each 16-lane half) -------
// quad_perm xor1 = 0xB1, quad_perm xor2 = 0x4E,
// row_half_mirror = 0x141 (xor4 once quads uniform),
// row_mirror = 0x140 (xor8 once 8-groups uniform)
template <int CTRL>
__device__ __forceinline__ float dpp_movf(float x) {
    int r = __builtin_amdgcn_update_dpp(0, __builtin_bit_cast(int, x),
                                        CTRL, 0xF, 0xF, true);
    return __builtin_bit_cast(float, r);
}
__device__ __forceinline__ float red16_max(float x) {
    x = fmaxf(x, dpp_movf<0xB1>(x));
    x = fmaxf(x, dpp_movf<0x4E>(x));
    x = fmaxf(x, dpp_movf<0x141>(x));
    x = fmaxf(x, dpp_movf<0x140>(x));
    return x;
}
__device__ __forceinline__ float red16_sum(float x) {
    x += dpp_movf<0xB1>(x);
    x += dpp_movf<0x4E>(x);
    x += dpp_movf<0x141>(x);
    x += dpp_movf<0x140>(x);
    return x;
}

// ---------------- Kernel 1: AdaptiveAvgPool2d(1) over prototypes ----------
__global__ void __launch_bounds__(256) pool_kernel(const float* __restrict__ proto,
                                                   float* __restrict__ ws) {
    __shared__ float red[256];
    const int m = blockIdx.x;
    const int tid = threadIdx.x;
    const float4* p = (const float4*)(proto + m * 25600);  // 6400 float4
    float s = 0.f;
    for (int i = tid; i < 6400; i += 256) {
        float4 v = p[i];
        s += (v.x + v.y) + (v.z + v.w);
    }
    red[tid] = s;
    __syncthreads();
    for (int off = 128; off > 0; off >>= 1) {
        if (tid < off) red[tid] += red[tid + off];
        __syncthreads();
    }
    if (tid == 0) ws[WS_POOLED + m] = red[0] * (1.f / 25600.f);
}

// ---------------- Kernel 2: fold K/V layernorms into Kw/Kb/Kc/Vo ----------
__global__ void __launch_bounds__(256) prep_kernel(
    const float* __restrict__ qw, const float* __restrict__ qb,
    const float* __restrict__ qg, const float* __restrict__ qbeta,
    const float* __restrict__ kw, const float* __restrict__ kb,
    const float* __restrict__ kg, const float* __restrict__ kbeta,
    const float* __restrict__ vw, const float* __restrict__ vb,
    const float* __restrict__ vg, const float* __restrict__ vbeta,
    const float* __restrict__ ow, float* __restrict__ ws) {
    __shared__ float s_wp[EE];   // (q_w - mw) * q_g
    __shared__ float s_bp[EE];   // (q_b - mb) * q_g
    __shared__ float s_qbe[EE];  // q_beta
    __shared__ float st[2];
    const int tid = threadIdx.x;
    if (tid == 0) {
        float mw = 0.f, mb = 0.f;
        for (int e = 0; e < EE; ++e) { mw += qw[e]; mb += qb[e]; }
        mw *= (1.f / EE); mb *= (1.f / EE);
        float Vw = 0.f, Cwb = 0.f, Vb = 0.f;
        for (int e = 0; e < EE; ++e) {
            float dw = qw[e] - mw, db = qb[e] - mb;
            Vw += dw * dw; Cwb += dw * db; Vb += db * db;
        }
        ws[WS_STATS + 0] = Vw * (1.f / EE);
        ws[WS_STATS + 1] = Cwb * (1.f / EE);
        ws[WS_STATS + 2] = Vb * (1.f / EE);
        st[0] = mw; st[1] = mb;
    }
    __syncthreads();
    if (tid < EE) {
        const float mw = st[0], mb = st[1];
        s_wp[tid]  = (qw[tid] - mw) * qg[tid];
        s_bp[tid]  = (qb[tid] - mb) * qg[tid];
        s_qbe[tid] = qbeta[tid];
    }
    __syncthreads();
    if (tid < NM) {
        const int m = tid;
        const float p = ws[WS_POOLED + m];
        float ks = 0.f, ksq = 0.f, vs = 0.f, vsq = 0.f;
        for (int e = 0; e < EE; ++e) {
            float x = p * kw[e] + kb[e]; ks += x; ksq += x * x;
            float y = p * vw[e] + vb[e]; vs += y; vsq += y * y;
        }
        const float kmu = ks * (1.f / EE), vmu = vs * (1.f / EE);
        const float krs = __builtin_amdgcn_rsqf(ksq * (1.f / EE) - kmu * kmu + LN_EPS);
        const float vrs = __builtin_amdgcn_rsqf(vsq * (1.f / EE) - vmu * vmu + LN_EPS);
        for (int h = 0; h < 4; ++h) {
            float aw = 0.f, ab = 0.f, ac = 0.f, av = 0.f;
            for (int d = 0; d < 32; ++d) {
                const int e = h * 32 + d;
                float kln = (p * kw[e] + kb[e] - kmu) * krs * kg[e] + kbeta[e];
                float vln = (p * vw[e] + vb[e] - vmu) * vrs * vg[e] + vbeta[e];
                aw += s_wp[e] * kln;
                ab += s_bp[e] * kln;
                ac += s_qbe[e] * kln;
                av += ow[e] * vln;
            }
            ws[WS_K + 0   + h * 64 + m] = aw;
            ws[WS_K + 256 + h * 64 + m] = ab;
            ws[WS_K + 512 + h * 64 + m] = ac;
            ws[WS_K + 768 + h * 64 + m] = av;
        }
    }
}

// ---------------- Kernel 3: fused WMMA attention + softmax + gate ---------
// Per wave: 16 (n,a) rows. Logits built with V_WMMA_F32_16X16X4_F32:
//   A (16x4): lanes 0-15 hold (K0=alpha, K1=beta), lanes 16-31 (K2=gamma, K3=0)
//   B (4x16): vgpr0 lanes 0-15 = Kw row (K0), lanes 16-31 = Kc row (K2);
//             vgpr1 lanes 0-15 = Kb row (K1), lanes 16-31 = 0   (K3)
//   D (16x16): lanes 0-15 rows 0-7 (vgpr=row), lanes 16-31 rows 8-15
__global__ void __launch_bounds__(256) attn_kernel(
    const float* __restrict__ coeff, const float* __restrict__ ws,
    const float* __restrict__ out_b_p, const float* __restrict__ gate_w,
    const float* __restrict__ gate_b_p, float* __restrict__ outp) {
    __shared__ float sK[1024];   // Kw | Kb | Kc | Vo
    __shared__ float sref[128];
    const int tid = threadIdx.x;
    for (int i = tid; i < 1024; i += 256) sK[i] = ws[WS_K + i];
    __syncthreads();

    const float Vw = ws[WS_STATS + 0];
    const float Cwb = ws[WS_STATS + 1];
    const float Vb = ws[WS_STATS + 2];

    const int wave = tid >> 5;
    const int lane = tid & 31;
    const int lane15 = lane & 15;
    const bool upper = lane >= 16;
    const int bofs = upper ? 512 : 0;   // lower half reads Kw, upper reads Kc

    const int tile = blockIdx.x * 8 + wave;       // < 33600, exact
    const int n = tile / TILES_PER_N;
    const int a0 = (tile - n * TILES_PER_N) * 16;
    const int base = n * AA + a0;

    const float c = coeff[base + lane15];
    const float rc = __builtin_amdgcn_rsqf(c * c * Vw + 2.f * c * Cwb + Vb + LN_EPS);
    v2f amat;
    amat.x = upper ? SCALE_QK : (SCALE_QK * c * rc);  // K0=alpha / K2=gamma
    amat.y = upper ? 0.f : (SCALE_QK * rc);           // K1=beta  / K3=0

    // preload the 16 Vo values this lane will need (one per chunk)
    float voA[16];
#pragma unroll
    for (int q = 0; q < 16; ++q) voA[q] = sK[768 + q * 16 + lane15];

    float acc[8];
#pragma unroll
    for (int j = 0; j < 8; ++j) acc[j] = 0.f;

#pragma unroll
    for (int h = 0; h < 4; ++h) {
        v8f L[4];
#pragma unroll
        for (int ch = 0; ch < 4; ++ch) {
            const int col = h * 64 + ch * 16 + lane15;
            const float bx = sK[bofs + col];        // Kw (lower) / Kc (upper)
            const float kbv = sK[256 + col];        // Kb
            v2f bmat;
            bmat.x = bx;
            bmat.y = upper ? 0.f : kbv;             // v_cndmask, no branch
            v8f z = {0.f, 0.f, 0.f, 0.f, 0.f, 0.f, 0.f, 0.f};
            L[ch] = __builtin_amdgcn_wmma_f32_16x16x4_f32(
                false, amat, false, bmat, (short)0, z, false, false);
        }
#pragma unroll
        for (int j = 0; j < 8; ++j) {
            float mx = fmaxf(fmaxf(L[0][j], L[1][j]), fmaxf(L[2][j], L[3][j]));
            mx = red16_max(mx);
            float S = 0.f, T = 0.f;
#pragma unroll
            for (int ch = 0; ch < 4; ++ch) {
                float e = __expf(L[ch][j] - mx);
                S += e;
                T += e * voA[h * 4 + ch];
            }
            S = red16_sum(S);
            T = red16_sum(T);
            acc[j] += T * __builtin_amdgcn_rcpf(S);  // per-head softmax dot
        }
    }

    const float ob = out_b_p[0];
    if (lane == 0) {
#pragma unroll
        for (int j = 0; j < 8; ++j) sref[wave * 16 + j] = acc[j] + ob;
    } else if (lane == 16) {
#pragma unroll
        for (int j = 0; j < 8; ++j) sref[wave * 16 + 8 + j] = acc[j] + ob;
    }
    __syncthreads();

    if (tid < 128) {
        const int tl = tid >> 4, r = tid & 15;
        const int tg = blockIdx.x * 8 + tl;
        const int nn = tg / TILES_PER_N;
        const int idx = nn * AA + (tg - nn * TILES_PER_N) * 16 + r;
        const float cc = coeff[idx];
        const float ref = sref[tid];
        const float x = cc * gate_w[0] + ref * gate_w[1] + gate_b_p[0];
        const float g = 1.f / (1.f + __expf(-x));
        outp[idx] = g * ref + (1.f - g) * cc;
    }
}

extern "C" void kernel_launch(void* const* d_in, const int* in_sizes, int n_in,
                              void* d_out, int out_size, void* d_ws, size_t ws_size,
                              hipStream_t stream) {
    const float* coeff  = (const float*)d_in[0];
    const float* proto  = (const float*)d_in[1];
    const float* q_w    = (const float*)d_in[2];
    const float* q_b    = (const float*)d_in[3];
    const float* q_g    = (const float*)d_in[4];
    const float* q_beta = (const float*)d_in[5];
    const float* k_w    = (const float*)d_in[6];
    const float* k_b    = (const float*)d_in[7];
    const float* k_g    = (const float*)d_in[8];
    const float* k_beta = (const float*)d_in[9];
    const float* v_w    = (const float*)d_in[10];
    const float* v_b    = (const float*)d_in[11];
    const float* v_g    = (const float*)d_in[12];
    const float* v_beta = (const float*)d_in[13];
    const float* out_w  = (const float*)d_in[14];
    const float* out_b  = (const float*)d_in[15];
    const float* gate_w = (const float*)d_in[16];
    const float* gate_b = (const float*)d_in[17];
    float* ws = (float*)d_ws;

    pool_kernel<<<NM, 256, 0, stream>>>(proto, ws);
    prep_kernel<<<1, 256, 0, stream>>>(q_w, q_b, q_g, q_beta,
                                       k_w, k_b, k_g, k_beta,
                                       v_w, v_b, v_g, v_beta,
                                       out_w, ws);
    attn_kernel<<<N_TILES / 8, 256, 0, stream>>>(coeff, ws, out_b, gate_w,
                                                 gate_b, (float*)d_out);
}